// TanhFixexPointLayer_17145509446156
// MI455X (gfx1250) — compile-verified
//
#include <hip/hip_runtime.h>

typedef __attribute__((ext_vector_type(2))) float v2f;
typedef __attribute__((ext_vector_type(8))) float v8f;

#define DIM        512          // feature dim D
#define BM         32           // rows per workgroup
#define LDS_PITCH  516          // padded row pitch (floats): 516 mod 64 = 4 -> conflict-free b64 loads
#define THREADS    128          // 4 waves (wave32)
#define ITERS      50

__device__ __forceinline__ float fast_tanh(float v) {
#if __has_builtin(__builtin_amdgcn_tanhf)
  return __builtin_amdgcn_tanhf(v);          // V_TANH_F32 (CDNA5 trans op)
#else
  // tanh(v) = 1 - 2/(exp(2v)+1); exp(2v) = exp2(2*log2(e)*v)
  float p = __builtin_amdgcn_exp2f(v * 2.885390081777927f);
  return 1.0f - 2.0f * __builtin_amdgcn_rcpf(p + 1.0f);
#endif
}

// Epilogue for one 16x16 f32 C/D tile: add x, tanh, store to LDS z-buffer.
// C/D layout: VGPR r, lane L -> (M = r + 8*(L/16), N = L%16)
__device__ __forceinline__ void tile_epilogue(const v8f& acc, int mt, int j,
                                              const float* __restrict__ x,
                                              float* __restrict__ zout,
                                              int m0, int half, int ln)
{
  #pragma unroll
  for (int r = 0; r < 8; ++r) {
    int ml = mt + r + 8 * half;                       // local row 0..31
    float v = acc[r] + x[(size_t)(m0 + ml) * DIM + (j + ln)];
    zout[ml * LDS_PITCH + (j + ln)] = fast_tanh(v);
  }
}

__launch_bounds__(THREADS)
__global__ void tanh_fixed_point_kernel(const float* __restrict__ x,
                                        const float* __restrict__ W,
                                        float* __restrict__ out)
{
  extern __shared__ float lds[];                 // 2 * BM * LDS_PITCH floats
  float* zb0 = lds;
  float* zb1 = lds + BM * LDS_PITCH;

  const int tid  = threadIdx.x;
  const int lane = tid & 31;
  const int wave = tid >> 5;       // 0..3, each wave owns 8 column tiles
  const int half = lane >> 4;      // 0 or 1
  const int ln   = lane & 15;
  const int m0   = blockIdx.x * BM;

  // ---- iteration 1: z = tanh(0 @ W^T + x) = tanh(x) ----
  for (int i = tid; i < BM * DIM; i += THREADS) {
    int r = i >> 9, c = i & (DIM - 1);
    zb0[r * LDS_PITCH + c] = fast_tanh(x[(size_t)(m0 + r) * DIM + c]);
  }
  __syncthreads();

  int cur = 0;
  for (int it = 1; it < ITERS; ++it) {
    const float* zin  = cur ? zb1 : zb0;
    float*       zout = cur ? zb0 : zb1;

    // A operand (16x4 f32): lane L, VGPR v -> (M = L%16, K = 2*(L/16)+v)
    const float* a0p = zin + ( 0 + ln) * LDS_PITCH + half * 2;   // rows 0..15
    const float* a1p = zin + (16 + ln) * LDS_PITCH + half * 2;   // rows 16..31

    #pragma unroll
    for (int g = 0; g < 2; ++g) {
      const int jb = wave * 128 + g * 64;   // first column of this 4-tile group
      // B operand (4x16 f32): lane L, VGPR v -> (K = 2*(L/16)+v, N = L%16)
      // B[k][n] = W^T[k][n] = W[(jb+n)*DIM + k]  (W row-major, contiguous in k)
      const float* b0p = W + (size_t)(jb +  0 + ln) * DIM + half * 2;
      const float* b1p = W + (size_t)(jb + 16 + ln) * DIM + half * 2;
      const float* b2p = W + (size_t)(jb + 32 + ln) * DIM + half * 2;
      const float* b3p = W + (size_t)(jb + 48 + ln) * DIM + half * 2;

      v8f acc00 = {}, acc01 = {}, acc02 = {}, acc03 = {};
      v8f acc10 = {}, acc11 = {}, acc12 = {}, acc13 = {};

      #pragma unroll 4
      for (int k = 0; k < DIM; k += 4) {
        v2f a0 = *(const v2f*)(a0p + k);
        v2f a1 = *(const v2f*)(a1p + k);
        v2f b0 = *(const v2f*)(b0p + k);
        v2f b1 = *(const v2f*)(b1p + k);
        v2f b2 = *(const v2f*)(b2p + k);
        v2f b3 = *(const v2f*)(b3p + k);
        acc00 = __builtin_amdgcn_wmma_f32_16x16x4_f32(false, a0, false, b0, 0, acc00, false, false);
        acc01 = __builtin_amdgcn_wmma_f32_16x16x4_f32(false, a0, false, b1, 0, acc01, false, false);
        acc02 = __builtin_amdgcn_wmma_f32_16x16x4_f32(false, a0, false, b2, 0, acc02, false, false);
        acc03 = __builtin_amdgcn_wmma_f32_16x16x4_f32(false, a0, false, b3, 0, acc03, false, false);
        acc10 = __builtin_amdgcn_wmma_f32_16x16x4_f32(false, a1, false, b0, 0, acc10, false, false);
        acc11 = __builtin_amdgcn_wmma_f32_16x16x4_f32(false, a1, false, b1, 0, acc11, false, false);
        acc12 = __builtin_amdgcn_wmma_f32_16x16x4_f32(false, a1, false, b2, 0, acc12, false, false);
        acc13 = __builtin_amdgcn_wmma_f32_16x16x4_f32(false, a1, false, b3, 0, acc13, false, false);
      }

      tile_epilogue(acc00,  0, jb +  0, x, zout, m0, half, ln);
      tile_epilogue(acc01,  0, jb + 16, x, zout, m0, half, ln);
      tile_epilogue(acc02,  0, jb + 32, x, zout, m0, half, ln);
      tile_epilogue(acc03,  0, jb + 48, x, zout, m0, half, ln);
      tile_epilogue(acc10, 16, jb +  0, x, zout, m0, half, ln);
      tile_epilogue(acc11, 16, jb + 16, x, zout, m0, half, ln);
      tile_epilogue(acc12, 16, jb + 32, x, zout, m0, half, ln);
      tile_epilogue(acc13, 16, jb + 48, x, zout, m0, half, ln);
    }
    __syncthreads();
    cur ^= 1;
  }

  // ---- write final z to global ----
  const float* zf = cur ? zb1 : zb0;
  for (int i = tid; i < BM * DIM; i += THREADS) {
    int r = i >> 9, c = i & (DIM - 1);
    out[(size_t)(m0 + r) * DIM + c] = zf[r * LDS_PITCH + c];
  }
}

extern "C" void kernel_launch(void* const* d_in, const int* in_sizes, int n_in,
                              void* d_out, int out_size, void* d_ws, size_t ws_size,
                              hipStream_t stream) {
  (void)in_sizes; (void)n_in; (void)d_ws; (void)ws_size; (void)out_size;
  const float* x = (const float*)d_in[0];   // [16384, 512] fp32
  const float* W = (const float*)d_in[1];   // [512, 512] fp32
  float* out = (float*)d_out;               // [16384, 512] fp32

  const int B = 16384;
  const size_t shmem = 2u * BM * LDS_PITCH * sizeof(float);   // 132096 B (<= 320KB WG limit)
  dim3 grid(B / BM);        // 512 workgroups
  dim3 block(THREADS);      // 128 threads = 4 waves
  tanh_fixed_point_kernel<<<grid, block, shmem, stream>>>(x, W, out);
}